// EleEmbedding_26963804684390
// MI455X (gfx1250) — compile-verified
//
#include <hip/hip_runtime.h>

// ---------------------------------------------------------------------------
// EleEmbedding on MI455X (gfx1250, wave32).
//
// Algebraic collapse:  dot_i = (a_i*(x_i.w1 + c1) + (x_i.w2 + c2)) / sqrt(D)
//   w1 = q_w^T @ k_w, w2 = q_w^T @ k_b, c1 = q_b.k_w, c2 = q_b.k_b,
//   a_i = ele[batch[i]];   out_i = softmax_seg(dot)_i * a_i * v_w.
// Traffic = 256MB (x read) + 256MB (out write) + ~6MB -> ~22us @ 23.3 TB/s.
// WMMA (V_WMMA_F32_16X16X4_F32) does the real matrix product q_w^T @ [k_w|k_b].
// ---------------------------------------------------------------------------

typedef __attribute__((ext_vector_type(2))) float v2f;
typedef __attribute__((ext_vector_type(8))) float v8f;

#define ND 128  // NODE_DIM, fixed by the reference

// ---------------------------------------------------------------------------
// Kernel 1: w1/w2 = q_w^T @ [k_w | k_b] via WMMA f32 16x16x4; c1/c2 dots.
// One block, 8 waves; wave w owns output rows [16w, 16w+16).
// A-fragment layout (ISA 7.12.2, 32-bit A 16x4): lanes 0-15 hold K=0,1 in
// v0,v1 for row M=lane; lanes 16-31 hold K=2,3 for row M=lane-16.
// B (4x16) / D (16x16) follow the row-striped pattern (VGPR r -> rows r, r+half).
// All loads are unconditional; B-fragment lane masking is done with selects
// (v_cndmask), so EXEC stays all-1s and no divergent clauses appear in the
// WMMA loop.
// ---------------------------------------------------------------------------
__global__ void precompute_wmma(const float* __restrict__ q_w, const float* __restrict__ q_b,
                                const float* __restrict__ k_w, const float* __restrict__ k_b,
                                float* __restrict__ w1, float* __restrict__ w2,
                                float* __restrict__ cc)
{
  const int lane = threadIdx.x & 31;
  const int wv   = threadIdx.x >> 5;   // 0..7 -> M tile
  const int m0   = wv * 16;
  const int half = lane >> 4;          // 0: lanes 0-15, 1: lanes 16-31
  const int lr   = lane & 15;

  const float sel_w = (lr == 0) ? 1.f : 0.f;   // column 0 of B = k_w
  const float sel_b = (lr == 1) ? 1.f : 0.f;   // column 1 of B = k_b

  v8f acc = {0.f, 0.f, 0.f, 0.f, 0.f, 0.f, 0.f, 0.f};

#pragma unroll 8
  for (int k0 = 0; k0 < ND; k0 += 4) {
    const int kx = k0 + 2 * half;                 // this lane's K pair: (kx, kx+1)
    // Unconditional scalar loads (same address across each 16-lane half).
    const float kw0 = k_w[kx], kw1 = k_w[kx + 1];
    const float kb0 = k_b[kx], kb1 = k_b[kx + 1];
    v2f a, b;
    // A[m][k] = (q_w^T)[m0+m][k0+k] = q_w[(k0+k)*ND + m0+m]
    a.x = q_w[(kx + 0) * ND + m0 + lr];
    a.y = q_w[(kx + 1) * ND + m0 + lr];
    // B[k][n]: n=0 -> k_w[k], n=1 -> k_b[k], else 0  (pure selects/FMAs)
    b.x = sel_w * kw0 + sel_b * kb0;
    b.y = sel_w * kw1 + sel_b * kb1;
    acc = __builtin_amdgcn_wmma_f32_16x16x4_f32(false, a, false, b, (short)0, acc,
                                                false, false);
  }

  // D 16x16 f32: VGPR r holds rows (r, r+8); lane column N = lr.
  const int mrow = m0 + 8 * half;
  if (lr == 0) {
#pragma unroll
    for (int r = 0; r < 8; ++r) w1[mrow + r] = acc[r];
  } else if (lr == 1) {
#pragma unroll
    for (int r = 0; r < 8; ++r) w2[mrow + r] = acc[r];
  }

  // c1 = q_b . k_w (wave 0), c2 = q_b . k_b (wave 1)
  if (threadIdx.x < 64) {
    const int    which = threadIdx.x >> 5;
    const float* kk    = which ? k_b : k_w;
    float s = 0.f;
    for (int j = lane; j < ND; j += 32) s += q_b[j] * kk[j];
#pragma unroll
    for (int off = 16; off >= 1; off >>= 1) s += __shfl_xor(s, off, 32);
    if (lane == 0) cc[which] = s;
  }
}

// ---------------------------------------------------------------------------
// Kernel 2: per-node dot. One wave per node row: 32 lanes x float4 = one 512B
// row of x, perfectly coalesced. Two independent rows in flight per iteration
// for ILP / latency hiding; xor-shuffle tree reduction; global_prefetch_b8
// warms the wave's next pair of rows.
// ---------------------------------------------------------------------------
__global__ void node_dot(const float* __restrict__ x, const float* __restrict__ ele,
                         const int* __restrict__ batch,
                         const float* __restrict__ w1, const float* __restrict__ w2,
                         const float* __restrict__ cc,
                         float* __restrict__ dot_out, int N)
{
  const int lane   = threadIdx.x & 31;
  const int wave   = blockIdx.x * (blockDim.x >> 5) + (threadIdx.x >> 5);
  const int nwaves = gridDim.x * (blockDim.x >> 5);

  const float4 w1v = reinterpret_cast<const float4*>(w1)[lane];
  const float4 w2v = reinterpret_cast<const float4*>(w2)[lane];
  const float  c1 = cc[0], c2 = cc[1];
  const float  scale = 0.08838834764831845f;  // 1/sqrt(128)

  int i = wave;
  // Main 2-row unrolled loop: rows i and i+nwaves processed with both loads
  // issued back-to-back before either reduction.
  for (; i + nwaves < N; i += 2 * nwaves) {
    const int j = i + nwaves;
    const float4 xv0 = reinterpret_cast<const float4*>(x + (size_t)i * ND)[lane];
    const float4 xv1 = reinterpret_cast<const float4*>(x + (size_t)j * ND)[lane];
    if (i + 2 * nwaves < N)  // wave-uniform; speculative prefetch of next rows
      __builtin_prefetch(x + (size_t)(i + 2 * (size_t)nwaves) * ND + lane * 4, 0, 0);
    if (j + 2 * nwaves < N)
      __builtin_prefetch(x + (size_t)(j + 2 * (size_t)nwaves) * ND + lane * 4, 0, 0);

    float d1a = xv0.x * w1v.x + xv0.y * w1v.y + xv0.z * w1v.z + xv0.w * w1v.w;
    float d2a = xv0.x * w2v.x + xv0.y * w2v.y + xv0.z * w2v.z + xv0.w * w2v.w;
    float d1b = xv1.x * w1v.x + xv1.y * w1v.y + xv1.z * w1v.z + xv1.w * w1v.w;
    float d2b = xv1.x * w2v.x + xv1.y * w2v.y + xv1.z * w2v.z + xv1.w * w2v.w;
#pragma unroll
    for (int off = 16; off >= 1; off >>= 1) {
      d1a += __shfl_xor(d1a, off, 32);
      d2a += __shfl_xor(d2a, off, 32);
      d1b += __shfl_xor(d1b, off, 32);
      d2b += __shfl_xor(d2b, off, 32);
    }
    if (lane == 0) {
      const int   bi = batch[i];
      const int   bj = batch[j];
      const float ai = ele[bi];
      const float aj = ele[bj];
      dot_out[i] = (ai * (d1a + c1) + (d2a + c2)) * scale;
      dot_out[j] = (aj * (d1b + c1) + (d2b + c2)) * scale;
    }
  }
  // Tail: at most one remaining row for this wave.
  if (i < N) {
    const float4 xv = reinterpret_cast<const float4*>(x + (size_t)i * ND)[lane];
    float d1 = xv.x * w1v.x + xv.y * w1v.y + xv.z * w1v.z + xv.w * w1v.w;
    float d2 = xv.x * w2v.x + xv.y * w2v.y + xv.z * w2v.z + xv.w * w2v.w;
#pragma unroll
    for (int off = 16; off >= 1; off >>= 1) {
      d1 += __shfl_xor(d1, off, 32);
      d2 += __shfl_xor(d2, off, 32);
    }
    if (lane == 0) {
      const int   b = batch[i];
      const float a = ele[b];
      dot_out[i] = (a * (d1 + c1) + (d2 + c2)) * scale;
    }
  }
}

// ---------------------------------------------------------------------------
// Kernel 3: deterministic segment softmax stats (batch is sorted, so each
// segment is a contiguous range found by binary search). One wave per segment:
// fixed-order max reduction, then fixed-order exp-sum. No atomics ->
// bitwise-reproducible across graph replays.
// ---------------------------------------------------------------------------
__device__ __forceinline__ int lower_bound_i(const int* __restrict__ a, int n, int key) {
  int lo = 0, hi = n;
  while (lo < hi) {
    const int mid = (lo + hi) >> 1;
    if (a[mid] < key) lo = mid + 1; else hi = mid;
  }
  return lo;
}

__global__ void segment_stats(const float* __restrict__ dot, const int* __restrict__ batch,
                              float* __restrict__ segmax, float* __restrict__ segsum,
                              int N, int B)
{
  const int lane   = threadIdx.x & 31;
  const int wave   = blockIdx.x * (blockDim.x >> 5) + (threadIdx.x >> 5);
  const int nwaves = gridDim.x * (blockDim.x >> 5);

  for (int b = wave; b < B; b += nwaves) {
    const int s0 = lower_bound_i(batch, N, b);      // uniform across the wave
    const int s1 = lower_bound_i(batch, N, b + 1);

    float mx = -__builtin_inff();
    for (int i = s0 + lane; i < s1; i += 32) mx = fmaxf(mx, dot[i]);
#pragma unroll
    for (int off = 16; off >= 1; off >>= 1) mx = fmaxf(mx, __shfl_xor(mx, off, 32));

    float ssum = 0.f;
    for (int i = s0 + lane; i < s1; i += 32) ssum += expf(dot[i] - mx);
#pragma unroll
    for (int off = 16; off >= 1; off >>= 1) ssum += __shfl_xor(ssum, off, 32);

    if (lane == 0) { segmax[b] = mx; segsum[b] = ssum; }
  }
}

// ---------------------------------------------------------------------------
// Kernel 4: out[i,:] = (exp(dot_i - max_b)/(sum_b+1e-16)) * ele[b] * v_w[:].
// One wave per node writes one coalesced 512B row. This is the 256MB store.
// ---------------------------------------------------------------------------
__global__ void scatter_out(const float* __restrict__ dot, const float* __restrict__ segmax,
                            const float* __restrict__ segsum, const int* __restrict__ batch,
                            const float* __restrict__ ele, const float* __restrict__ v_w,
                            float* __restrict__ out, int N)
{
  const int lane   = threadIdx.x & 31;
  const int wave   = blockIdx.x * (blockDim.x >> 5) + (threadIdx.x >> 5);
  const int nwaves = gridDim.x * (blockDim.x >> 5);

  const float4 vw = reinterpret_cast<const float4*>(v_w)[lane];

  for (int i = wave; i < N; i += nwaves) {
    const int   b = batch[i];
    const float e = expf(dot[i] - segmax[b]);
    const float s = e / (segsum[b] + 1e-16f) * ele[b];
    float4 o;
    o.x = s * vw.x; o.y = s * vw.y; o.z = s * vw.z; o.w = s * vw.w;
    reinterpret_cast<float4*>(out + (size_t)i * ND)[lane] = o;
  }
}

// ---------------------------------------------------------------------------
extern "C" void kernel_launch(void* const* d_in, const int* in_sizes, int n_in,
                              void* d_out, int out_size, void* d_ws, size_t ws_size,
                              hipStream_t stream)
{
  const float* x     = (const float*)d_in[0];
  const float* ele   = (const float*)d_in[1];
  const float* q_w   = (const float*)d_in[2];
  const float* q_b   = (const float*)d_in[3];
  const float* k_w   = (const float*)d_in[4];
  const float* k_b   = (const float*)d_in[5];
  const float* v_w   = (const float*)d_in[6];
  const int*   batch = (const int*)d_in[7];   // JAX default x64-off -> int32

  const int N = in_sizes[0] / ND;   // 500000
  const int B = in_sizes[1];        // 4096
  float* out = (float*)d_out;

  // Workspace carve-up (256B-aligned regions)
  char* p = (char*)d_ws;
  auto take = [&](size_t bytes) { char* r = p; p += (bytes + 255) & ~(size_t)255; return r; };
  float* dot    = (float*)take((size_t)N * sizeof(float));
  float* segmax = (float*)take((size_t)B * sizeof(float));
  float* segsum = (float*)take((size_t)B * sizeof(float));
  float* w1     = (float*)take(ND * sizeof(float));
  float* w2     = (float*)take(ND * sizeof(float));
  float* cc     = (float*)take(2 * sizeof(float));

  precompute_wmma<<<1, 256, 0, stream>>>(q_w, q_b, k_w, k_b, w1, w2, cc);

  node_dot<<<2048, 256, 0, stream>>>(x, ele, batch, w1, w2, cc, dot, N);

  segment_stats<<<512, 256, 0, stream>>>(dot, batch, segmax, segsum, N, B);

  scatter_out<<<2048, 256, 0, stream>>>(dot, segmax, segsum, batch, ele, v_w, out, N);
}